// CombinedRankingLoss_39350490366203
// MI455X (gfx1250) — compile-verified
//
#include <hip/hip_runtime.h>
#include <stdint.h>
#include <math.h>

#define N_ELEM 1024
#define TPB    256
#define EPW    4   // elements per thread (N_ELEM / TPB)

__device__ __forceinline__ float lse2(float a, float b) {
    // logaddexp; safe when exactly one side is -inf
    float m = fmaxf(a, b);
    float d = fminf(a, b) - m;          // <= 0, may be -inf
    return m + log1pf(expf(d));
}

__global__ __launch_bounds__(TPB)
void row_loss_kernel(const float* __restrict__ scores,
                     const int*   __restrict__ labels,
                     const float* __restrict__ temperature,
                     float*       __restrict__ row_out /* [B*4]: listmle, focal_sum, smooth_sum */)
{
    __shared__ float              s_scores[N_ELEM];   // 4 KB
    __shared__ int                s_labels[N_ELEM];   // 4 KB
    __shared__ float              s_perm[N_ELEM];     // 4 KB (reversed-sorted calibrated scores)
    __shared__ unsigned long long s_scanu[TPB];       // 2 KB
    __shared__ float              s_scanf[TPB];       // 1 KB
    __shared__ float              s_red[TPB * 4];     // 4 KB

    const int t   = threadIdx.x;
    const int row = blockIdx.x;
    const size_t rowbase = (size_t)row * N_ELEM;

    // ---- CDNA5 async DMA: copy scores+labels row into LDS (b128 per thread per array) ----
    {
        unsigned long long gs = (unsigned long long)(scores + rowbase + t * EPW);
        unsigned long long gl = (unsigned long long)(labels + rowbase + t * EPW);
        unsigned int lds_s = (unsigned int)(uintptr_t)(&s_scores[t * EPW]);
        unsigned int lds_l = (unsigned int)(uintptr_t)(&s_labels[t * EPW]);
        asm volatile("global_load_async_to_lds_b128 %0, %1, off"
                     :: "v"(lds_s), "v"(gs) : "memory");
        asm volatile("global_load_async_to_lds_b128 %0, %1, off"
                     :: "v"(lds_l), "v"(gl) : "memory");
        asm volatile("s_wait_asynccnt 0" ::: "memory");
    }
    __syncthreads();

    float temp = temperature[0];
    temp = fminf(fmaxf(temp, 0.1f), 5.0f);
    const float inv_temp = 1.0f / temp;

    // ---- elementwise losses + per-thread label histogram (original order) ----
    float xv[EPW]; int lv[EPW];
    int   cnt[5] = {0, 0, 0, 0, 0};
    float focal_acc = 0.f, smooth_acc = 0.f, spos_acc = 0.f;
    #pragma unroll
    for (int k = 0; k < EPW; ++k) {
        float x = s_scores[t * EPW + k] * inv_temp;
        int   l = s_labels[t * EPW + k];
        l = (l < 0) ? 0 : ((l > 4) ? 4 : l);
        xv[k] = x; lv[k] = l;
        cnt[l]++;
        float b   = (l > 0) ? 1.f : 0.f;
        float ax  = fabsf(x);
        float e   = expf(-ax);
        float l1p = log1pf(e);
        float mx  = fmaxf(x, 0.f);
        float ce  = mx - x * b + l1p;               // BCE with binary target
        float den = 1.f + e;
        float p   = (x >= 0.f) ? (1.f / den) : (e / den);   // sigmoid(x)
        float pt  = b * p + (1.f - b) * (1.f - p);
        float omt = 1.f - pt;
        focal_acc  += 0.25f * omt * omt * ce;                // alpha=0.25, gamma=2
        float sl    = 0.9f * b + (0.1f / (float)N_ELEM);     // label smoothing
        smooth_acc += mx - x * sl + l1p;
        spos_acc   += b * x;                                 // sum of scores with label>0
    }

    // ---- stable counting sort ranks: one u64 scan carries all 5 label counters ----
    unsigned long long packed = 0ull;
    #pragma unroll
    for (int l = 0; l < 5; ++l) packed |= ((unsigned long long)cnt[l]) << (12 * l);
    s_scanu[t] = packed;
    __syncthreads();
    for (int off = 1; off < TPB; off <<= 1) {
        unsigned long long add = (t >= off) ? s_scanu[t - off] : 0ull;
        __syncthreads();
        s_scanu[t] += add;
        __syncthreads();
    }
    const unsigned long long incl = s_scanu[t];
    const unsigned long long tot  = s_scanu[TPB - 1];
    int excl[5], C[5];
    #pragma unroll
    for (int l = 0; l < 5; ++l) {
        C[l]    = (int)((tot  >> (12 * l)) & 0xFFFull);
        excl[l] = (int)((incl >> (12 * l)) & 0xFFFull) - cnt[l];
    }
    int base[5];
    base[0] = 0;
    #pragma unroll
    for (int l = 1; l < 5; ++l) base[l] = base[l - 1] + C[l - 1];
    const int C0 = C[0];

    // ---- scatter into reversed-sorted order:
    // sorted = stable desc-label; reversed = asc label, reverse original index within label
    {
        int seen[5] = {0, 0, 0, 0, 0};
        #pragma unroll
        for (int k = 0; k < EPW; ++k) {
            int l = lv[k];
            int r = excl[l] + seen[l];             // stable rank within label l
            seen[l]++;
            int pos = base[l] + (C[l] - 1 - r);    // reverse within label
            s_perm[pos] = xv[k];
        }
    }
    __syncthreads();

    // ---- inclusive prefix logaddexp scan over s_perm (== suffix LSE of sorted order) ----
    float a[EPW];
    a[0] = s_perm[t * EPW + 0];
    #pragma unroll
    for (int k = 1; k < EPW; ++k) a[k] = lse2(a[k - 1], s_perm[t * EPW + k]);
    s_scanf[t] = a[EPW - 1];
    __syncthreads();
    for (int off = 1; off < TPB; off <<= 1) {
        float v = (t >= off) ? s_scanf[t - off] : -__builtin_inff();
        __syncthreads();
        float cur = s_scanf[t];
        s_scanf[t] = lse2(cur, v);
        __syncthreads();
    }
    const float eprev = (t > 0) ? s_scanf[t - 1] : -__builtin_inff();
    float sumP = 0.f;
    #pragma unroll
    for (int k = 0; k < EPW; ++k) {
        float P = lse2(eprev, a[k]);
        if (t * EPW + k >= C0) sumP += P;          // positions with label > 0
    }

    // ---- deterministic block reduction of 4 partials ----
    s_red[t * 4 + 0] = focal_acc;
    s_red[t * 4 + 1] = smooth_acc;
    s_red[t * 4 + 2] = spos_acc;
    s_red[t * 4 + 3] = sumP;
    __syncthreads();
    for (int s = TPB / 2; s > 0; s >>= 1) {
        if (t < s) {
            s_red[t * 4 + 0] += s_red[(t + s) * 4 + 0];
            s_red[t * 4 + 1] += s_red[(t + s) * 4 + 1];
            s_red[t * 4 + 2] += s_red[(t + s) * 4 + 2];
            s_red[t * 4 + 3] += s_red[(t + s) * 4 + 3];
        }
        __syncthreads();
    }
    if (t == 0) {
        float K = (float)(N_ELEM - C0);
        float listmle = -(s_red[2] - s_red[3]) / (K + 1e-10f);
        row_out[(size_t)row * 4 + 0] = listmle;
        row_out[(size_t)row * 4 + 1] = s_red[0];
        row_out[(size_t)row * 4 + 2] = s_red[1];
    }
}

__global__ __launch_bounds__(TPB)
void finalize_kernel(const float* __restrict__ row_out, float* __restrict__ out, int B)
{
    __shared__ float red[TPB * 3];
    int t = threadIdx.x;
    float l = 0.f, f = 0.f, s = 0.f;
    for (int r = t; r < B; r += TPB) {
        l += row_out[(size_t)r * 4 + 0];
        f += row_out[(size_t)r * 4 + 1];
        s += row_out[(size_t)r * 4 + 2];
    }
    red[t * 3 + 0] = l; red[t * 3 + 1] = f; red[t * 3 + 2] = s;
    __syncthreads();
    for (int st = TPB / 2; st > 0; st >>= 1) {
        if (t < st) {
            red[t * 3 + 0] += red[(t + st) * 3 + 0];
            red[t * 3 + 1] += red[(t + st) * 3 + 1];
            red[t * 3 + 2] += red[(t + st) * 3 + 2];
        }
        __syncthreads();
    }
    if (t == 0) {
        float invB  = 1.0f / (float)B;
        float invBN = 1.0f / ((float)B * (float)N_ELEM);
        out[0] = 0.7f * red[0] * invB + 0.3f * red[1] * invBN + 0.1f * red[2] * invBN;
    }
}

extern "C" void kernel_launch(void* const* d_in, const int* in_sizes, int n_in,
                              void* d_out, int out_size, void* d_ws, size_t ws_size,
                              hipStream_t stream) {
    const float* scores      = (const float*)d_in[0];
    const float* temperature = (const float*)d_in[1];
    const int*   labels      = (const int*)d_in[2];
    const int B = in_sizes[0] / N_ELEM;

    float* row_out = (float*)d_ws;   // B * 4 floats of scratch

    row_loss_kernel<<<B, TPB, 0, stream>>>(scores, labels, temperature, row_out);
    finalize_kernel<<<1, TPB, 0, stream>>>(row_out, (float*)d_out, B);
}